// MetropolisHasting_37314675868413
// MI455X (gfx1250) — compile-verified
//
#include <hip/hip_runtime.h>
#include <stdint.h>

// ---------------------------------------------------------------------------
// RBM Metropolis-Hastings on MI455X (gfx1250, wave32)
//
// Phase 1 (WMMA): h = curr @ W + b           [4096 x 1024] f32, V_WMMA_F32_16X16X4_F32
// Phase 2 (MCMC): 132 sequential single-spin-flip steps, one block per chain,
//                 h kept in registers (4 f32/thread), spins in LDS,
//                 rank-1 incremental update instead of per-step GEMMs.
// ---------------------------------------------------------------------------

#define NCHAINS 4096
#define STATE   256
#define HIDDEN  1024
#define MH_ITERS 132
#define BURN    128

typedef __attribute__((ext_vector_type(2))) float v2f;
typedef __attribute__((ext_vector_type(8))) float v8f;

// ---------------------------------------------------------------------------
// Phase 1: h = A(4096x256) * W(256x1024) + b, f32 WMMA 16x16x4.
// One wave per 16x16 output tile; 4 waves per block.
//
// A-fragment layout (32-bit A 16x4, ISA 7.12.2):
//   lanes 0-15  : row M=lane,    v0 = K=k0+0, v1 = K=k0+1
//   lanes 16-31 : row M=lane-16, v0 = K=k0+2, v1 = K=k0+3
// B-fragment (4x16, row striped across lanes per VGPR):
//   v0: lanes 0-15 -> row k0+0, lanes 16-31 -> row k0+2
//   v1: lanes 0-15 -> row k0+1, lanes 16-31 -> row k0+3
// C/D: VGPR r -> M = r (lanes 0-15), M = 8+r (lanes 16-31), N = lane%16.
// ---------------------------------------------------------------------------
__global__ __launch_bounds__(128) void rbm_init_h_wmma(
    const float* __restrict__ A,     // curr  [4096 x 256]
    const float* __restrict__ W,     // W     [256 x 1024]
    const float* __restrict__ bvec,  // b     [1024]
    float* __restrict__ H)           // out h [4096 x 1024]
{
    const int lane = threadIdx.x & 31;
    const int wave = threadIdx.x >> 5;
    const int m0   = blockIdx.y * 16;                 // 0..4080
    const int n0   = (blockIdx.x * 4 + wave) * 16;    // 0..1008
    const int half = lane >> 4;                       // 0 | 1
    const int l    = lane & 15;

    v8f acc = {};
    const float* arow = A + (size_t)(m0 + l) * STATE + 2 * half;

    #pragma unroll 4
    for (int k0 = 0; k0 < STATE; k0 += 4) {
        v2f afrag, bfrag;
        afrag.x = arow[k0];
        afrag.y = arow[k0 + 1];
        const int kb = k0 + 2 * half;
        bfrag.x = W[(size_t)kb * HIDDEN + n0 + l];
        bfrag.y = W[(size_t)(kb + 1) * HIDDEN + n0 + l];
        acc = __builtin_amdgcn_wmma_f32_16x16x4_f32(
            /*neg_a=*/false, afrag, /*neg_b=*/false, bfrag,
            /*c_mod=*/(short)0, acc, /*reuse_a=*/false, /*reuse_b=*/false);
    }

    const int   ncol = n0 + l;
    const float bj   = bvec[ncol];
    float* hp = H + (size_t)(m0 + 8 * half) * HIDDEN + ncol;
    #pragma unroll
    for (int r = 0; r < 8; ++r)
        hp[(size_t)r * HIDDEN] = acc[r] + bj;
}

// ---------------------------------------------------------------------------
// Phase 2: sequential MH chain, one block (256 threads) per chain.
// ---------------------------------------------------------------------------
__device__ __forceinline__ uint32_t hash_u32(uint32_t a) {
    a ^= a >> 16; a *= 0x7feb352du;
    a ^= a >> 15; a *= 0x846ca68bu;
    a ^= a >> 16;
    return a;
}

// logcosh up to an additive constant (the -log2 cancels in the difference)
__device__ __forceinline__ float lc(float x) {
    float a = fabsf(x);
    return a + __logf(1.0f + __expf(-2.0f * a));
}

__global__ __launch_bounds__(256) void rbm_mcmc(
    const float*  __restrict__ curr,   // [4096 x 256] (never mutated)
    const float4* __restrict__ W4,     // W as [256 x 256] float4 rows
    const int*    __restrict__ seed_p,
    const float*  __restrict__ H,      // initial h from phase 1
    float*        __restrict__ out)    // [16384 x 256]
{
    const int c = blockIdx.x;
    const int t = threadIdx.x;

    __shared__ float s_spins[STATE];
    __shared__ float s_red[8];

    const uint32_t seed   = (uint32_t)seed_p[0];
    const uint32_t cmix   = 0x85EBCA6Bu * (uint32_t)c;

    s_spins[t] = curr[(size_t)c * STATE + t];
    float4 h = reinterpret_cast<const float4*>(H + (size_t)c * HIDDEN)[t];
    __syncthreads();

    for (int it = 0; it < MH_ITERS; ++it) {
        // counter-based RNG: deterministic in (seed, it, c)
        uint32_t r0  = hash_u32(hash_u32(seed + 0x9E3779B9u * (uint32_t)it) + cmix);
        uint32_t r1  = hash_u32(r0 ^ 0x68E31DA4u);
        const int   idx = (int)(r0 & (STATE - 1));
        const float u   = (float)(r1 >> 8) * (1.0f / 16777216.0f);

        const float s = s_spins[idx];
        const float4 w = W4[(size_t)idx * (HIDDEN / 4) + t];
        const float m = -2.0f * s;
        float4 d;
        d.x = m * w.x; d.y = m * w.y; d.z = m * w.z; d.w = m * w.w;

        float local = (lc(h.x + d.x) - lc(h.x))
                    + (lc(h.y + d.y) - lc(h.y))
                    + (lc(h.z + d.z) - lc(h.z))
                    + (lc(h.w + d.w) - lc(h.w));

        // prefetch next iteration's W row (RNG lets us look ahead)
        uint32_t n0 = hash_u32(hash_u32(seed + 0x9E3779B9u * (uint32_t)(it + 1)) + cmix);
        __builtin_prefetch(&W4[(size_t)(n0 & (STATE - 1)) * (HIDDEN / 4) + t], 0, 1);

        // block reduction of local over 8 waves
        float v = local;
        #pragma unroll
        for (int off = 16; off > 0; off >>= 1)
            v += __shfl_xor(v, off, 32);
        __syncthreads();                       // s_red safe to rewrite
        if ((t & 31) == 0) s_red[t >> 5] = v;
        __syncthreads();
        float dlp = s_red[0] + s_red[1] + s_red[2] + s_red[3]
                  + s_red[4] + s_red[5] + s_red[6] + s_red[7];

        // probs = min(exp(2*dlp), 1) == exp(min(2*dlp, 0))
        const float pr = __expf(fminf(2.0f * dlp, 0.0f));
        const bool accept = (u < pr);

        if (accept) { h.x += d.x; h.y += d.y; h.z += d.z; h.w += d.w; }

        __syncthreads();                       // all reads of s_spins done
        if (accept && t == 0) s_spins[idx] = -s;
        __syncthreads();                       // flip visible

        if (it >= BURN) {
            out[((size_t)c * 4 + (size_t)(it - BURN)) * STATE + t] = s_spins[t];
        }
    }
}

// ---------------------------------------------------------------------------
extern "C" void kernel_launch(void* const* d_in, const int* in_sizes, int n_in,
                              void* d_out, int out_size, void* d_ws, size_t ws_size,
                              hipStream_t stream) {
    const float* curr = (const float*)d_in[0];   // [4096 x 256]
    const float* W    = (const float*)d_in[1];   // [256 x 1024]
    const float* bvec = (const float*)d_in[2];   // [1024]
    const int*   seed = (const int*)d_in[3];     // scalar
    float* out = (float*)d_out;                  // [16384 x 256]
    float* H   = (float*)d_ws;                   // 4096*1024*4 = 16 MB scratch

    dim3 g1(HIDDEN / 64, NCHAINS / 16);          // 4 N-tiles per block (4 waves)
    rbm_init_h_wmma<<<g1, 128, 0, stream>>>(curr, W, bvec, H);

    rbm_mcmc<<<NCHAINS, STATE, 0, stream>>>(curr, (const float4*)W, seed, H, out);
}